// PennyLanePQC_21320217657899
// MI455X (gfx1250) — compile-verified
//
#include <hip/hip_runtime.h>
#include <hip/hip_bf16.h>
#include <math.h>

typedef __attribute__((ext_vector_type(2))) float v2f;
typedef __attribute__((ext_vector_type(8))) float v8f;

#define PI_HALF 1.57079632679489662f

// ---------------------------------------------------------------------------
// Phase 1: per-batch-element quantum circuit simulation in LDS.
// One workgroup (256 threads) per batch element; state (<=512 complex) in LDS.
// ---------------------------------------------------------------------------

__device__ inline void gate_rz(float* sr, float* si, int n, int wire, float theta) {
    float sv, cv;
    __sincosf(0.5f * theta, &sv, &cv);
    const int dim = 1 << n;
    const int bit = 1 << (n - 1 - wire);
    for (int i = threadIdx.x; i < dim; i += blockDim.x) {
        float sg = (i & bit) ? sv : -sv;   // e^{+-i theta/2}
        float re = sr[i], im = si[i];
        sr[i] = cv * re - sg * im;
        si[i] = cv * im + sg * re;
    }
    __syncthreads();
}

__device__ inline void gate_ry(float* sr, float* si, int n, int wire, float theta) {
    float sv, cv;
    __sincosf(0.5f * theta, &sv, &cv);
    const int dim = 1 << n;
    const int bit = 1 << (n - 1 - wire);
    const int half = dim >> 1;
    for (int p = threadIdx.x; p < half; p += blockDim.x) {
        int low = p & (bit - 1);
        int i0 = ((p ^ low) << 1) | low;
        int i1 = i0 | bit;
        float a0r = sr[i0], a0i = si[i0], a1r = sr[i1], a1i = si[i1];
        sr[i0] = cv * a0r - sv * a1r;  si[i0] = cv * a0i - sv * a1i;
        sr[i1] = sv * a0r + cv * a1r;  si[i1] = sv * a0i + cv * a1i;
    }
    __syncthreads();
}

__device__ inline void gate_rx(float* sr, float* si, int n, int wire, float theta) {
    float sv, cv;
    __sincosf(0.5f * theta, &sv, &cv);
    const int dim = 1 << n;
    const int bit = 1 << (n - 1 - wire);
    const int half = dim >> 1;
    for (int p = threadIdx.x; p < half; p += blockDim.x) {
        int low = p & (bit - 1);
        int i0 = ((p ^ low) << 1) | low;
        int i1 = i0 | bit;
        float a0r = sr[i0], a0i = si[i0], a1r = sr[i1], a1i = si[i1];
        // [[c, -i s],[-i s, c]]
        sr[i0] = cv * a0r + sv * a1i;   si[i0] = cv * a0i - sv * a1r;
        sr[i1] = sv * a0i + cv * a1r;   si[i1] = -sv * a0r + cv * a1i;
    }
    __syncthreads();
}

__device__ inline void gate_h(float* sr, float* si, int n, int wire) {
    const float r = 0.70710678118654752f;
    const int dim = 1 << n;
    const int bit = 1 << (n - 1 - wire);
    const int half = dim >> 1;
    for (int p = threadIdx.x; p < half; p += blockDim.x) {
        int low = p & (bit - 1);
        int i0 = ((p ^ low) << 1) | low;
        int i1 = i0 | bit;
        float a0r = sr[i0], a0i = si[i0], a1r = sr[i1], a1i = si[i1];
        sr[i0] = r * (a0r + a1r);  si[i0] = r * (a0i + a1i);
        sr[i1] = r * (a0r - a1r);  si[i1] = r * (a0i - a1i);
    }
    __syncthreads();
}

__device__ inline void gate_cnot(float* sr, float* si, int n, int c, int t) {
    const int dim = 1 << n;
    const int cbit = 1 << (n - 1 - c);
    const int tbit = 1 << (n - 1 - t);
    const int lo = (cbit < tbit) ? cbit : tbit;
    const int hi = cbit ^ tbit ^ lo;
    const int quarter = dim >> 2;
    for (int q = threadIdx.x; q < quarter; q += blockDim.x) {
        int x = q;
        int l1 = x & (lo - 1); x = ((x ^ l1) << 1) | l1;
        int l2 = x & (hi - 1); x = ((x ^ l2) << 1) | l2;
        int i0 = x | cbit;          // control = 1, target = 0
        int i1 = i0 | tbit;         // control = 1, target = 1
        float tr = sr[i0]; sr[i0] = sr[i1]; sr[i1] = tr;
        float ti = si[i0]; si[i0] = si[i1]; si[i1] = ti;
    }
    __syncthreads();
}

__device__ inline void gate_cz(float* sr, float* si, int n, int c, int t) {
    const int dim = 1 << n;
    const int cbit = 1 << (n - 1 - c);
    const int tbit = 1 << (n - 1 - t);
    const int lo = (cbit < tbit) ? cbit : tbit;
    const int hi = cbit ^ tbit ^ lo;
    const int quarter = dim >> 2;
    for (int q = threadIdx.x; q < quarter; q += blockDim.x) {
        int x = q;
        int l1 = x & (lo - 1); x = ((x ^ l1) << 1) | l1;
        int l2 = x & (hi - 1); x = ((x ^ l2) << 1) | l2;
        int i = x | cbit | tbit;
        sr[i] = -sr[i];
        si[i] = -si[i];
    }
    __syncthreads();
}

__device__ inline void conv_unit(float* sr, float* si, int n,
                                 float p0, float p1, float p2, int w0, int w1) {
    gate_rz(sr, si, n, w1, -PI_HALF);
    gate_cnot(sr, si, n, w1, w0);
    gate_rz(sr, si, n, w0, -2.f * p0 + PI_HALF);
    gate_ry(sr, si, n, w1, -PI_HALF + 2.f * p1);
    gate_cnot(sr, si, n, w0, w1);
    gate_ry(sr, si, n, w1, -2.f * p2 + PI_HALF);
    gate_cnot(sr, si, n, w1, w0);
    gate_rz(sr, si, n, w0, PI_HALF);
}

__device__ inline void layer_ansatz(float* sr, float* si, int n, const float* lp) {
    int ptr = 0;
    for (int i = 0; i + 1 < n; i += 2) {           // even pairs
        conv_unit(sr, si, n, lp[ptr], lp[ptr + 1], lp[ptr + 2], i, (i + 1) % n);
        ptr += 3;
    }
    for (int i = 1; i < n; i += 2) {               // odd pairs (wraps)
        conv_unit(sr, si, n, lp[ptr], lp[ptr + 1], lp[ptr + 2], i, (i + 1) % n);
        ptr += 3;
    }
    for (int w = 0; w < n; ++w) gate_h(sr, si, n, w);
    for (int i = 0; i + 1 < n; ++i) gate_cz(sr, si, n, i, i + 1);
    for (int i = 0; i < n; ++i) gate_rx(sr, si, n, i, lp[ptr + i]);
}

__device__ inline float block_norm(const float* sr, const float* si, float* red, int dim) {
    float s = 0.f;
    for (int i = threadIdx.x; i < dim; i += blockDim.x)
        s += sr[i] * sr[i] + si[i] * si[i];
    red[threadIdx.x] = s;
    __syncthreads();
    for (int off = 128; off > 0; off >>= 1) {
        if ((int)threadIdx.x < off) red[threadIdx.x] += red[threadIdx.x + off];
        __syncthreads();
    }
    float n = sqrtf(red[0]);
    __syncthreads();
    return n;
}

__global__ __launch_bounds__(256)
void pqc_circuit_kernel(const float* __restrict__ inp,
                        const float* __restrict__ aw,
                        float* __restrict__ XR, float* __restrict__ XI) {
    __shared__ float sr[512];
    __shared__ float si[512];
    __shared__ float red[256];
    const int b = blockIdx.x;
    const int t = threadIdx.x;

    sr[t] = inp[b * 256 + t];
    si[t] = 0.f;
    sr[256 + t] = 0.f;
    si[256 + t] = 0.f;
    __syncthreads();

    const float* wrow = aw + b * 194;

    // ansatz 1 (8 qubits), weights [0,64)
    for (int d = 0; d < 2; ++d) layer_ansatz(sr, si, 8, wrow + d * 32);
    // extend with zeros (upper half already zero), ansatz 2 (9 qubits), weights [64,130)
    for (int d = 0; d < 2; ++d) layer_ansatz(sr, si, 9, wrow + 64 + d * 33);

    // truncate to 256, normalize (+eps), leaky-relu on re/im, renormalize
    float nrm = block_norm(sr, si, red, 256);
    float inv = 1.f / (nrm + 1e-8f);
    {
        float a = sr[t] * inv, c = si[t] * inv;
        a = (a >= 0.f) ? a : 0.01f * a;
        c = (c >= 0.f) ? c : 0.01f * c;
        sr[t] = a; si[t] = c;
    }
    __syncthreads();
    nrm = block_norm(sr, si, red, 256);
    inv = 1.f / nrm;
    sr[t] *= inv; si[t] *= inv;
    __syncthreads();

    // ansatz 3 (8 qubits), weights [130,194)
    for (int d = 0; d < 2; ++d) layer_ansatz(sr, si, 8, wrow + 130 + d * 32);

    XR[b * 256 + t] = sr[t];
    XI[b * 256 + t] = si[t];
}

// ---------------------------------------------------------------------------
// Phase 2: build Hermitian planes Hr/Hi from ano_weights (C + C^dagger)/2.
// ano: [64][256][256][2] interleaved re/im.
// ---------------------------------------------------------------------------
__global__ __launch_bounds__(256)
void prep_H_kernel(const float* __restrict__ ano,
                   float* __restrict__ HR, float* __restrict__ HI) {
    const size_t idx = (size_t)blockIdx.x * blockDim.x + threadIdx.x; // < 64*256*256
    const int k = (int)(idx >> 16);
    const int i = (int)((idx >> 8) & 255);
    const int j = (int)(idx & 255);
    const size_t ij = (((size_t)k << 16) + ((size_t)i << 8) + j) * 2;
    const size_t ji = (((size_t)k << 16) + ((size_t)j << 8) + i) * 2;
    HR[idx] = 0.5f * (ano[ij]     + ano[ji]);
    HI[idx] = 0.5f * (ano[ij + 1] - ano[ji + 1]);
}

// ---------------------------------------------------------------------------
// Phase 3: out[b,k] = Re(psi^dag H_k psi) via f32 WMMA (16x16x4).
// Block = 128 threads (4 waves). Each block: one k, one 16-batch tile (in LDS,
// stride 257 to avoid bank conflicts). Waves split the i-range (64 rows each).
// Complex GEMM = 4 real WMMA accumulations, combined with VALU.
// ---------------------------------------------------------------------------
__global__ __launch_bounds__(128)
void expect_kernel(const float* __restrict__ XR, const float* __restrict__ XI,
                   const float* __restrict__ HR, const float* __restrict__ HI,
                   float* __restrict__ out) {
    __shared__ float lxr[16 * 257];
    __shared__ float lxi[16 * 257];
    __shared__ float red[4][16];

    const int k     = blockIdx.x >> 7;           // 64 hamiltonians
    const int bbase = (blockIdx.x & 127) * 16;   // 128 batch tiles of 16
    const int wave  = threadIdx.x >> 5;
    const int lane  = threadIdx.x & 31;

    // load psi tile [16 b][256 j] planar into LDS (padded stride 257)
    for (int e = threadIdx.x; e < 16 * 256; e += 128) {
        int row = e >> 8, colj = e & 255;
        lxr[row * 257 + colj] = XR[(bbase + row) * 256 + colj];
        lxi[row * 257 + colj] = XI[(bbase + row) * 256 + colj];
    }
    __syncthreads();

    const int col = lane & 15;            // N (batch) index / A-matrix row
    const int kk  = (lane >> 4) << 1;     // K pair selected by half-wave
    const float* HRk = HR + ((size_t)k << 16);
    const float* HIk = HI + ((size_t)k << 16);
    const float* bxr = lxr + col * 257;
    const float* bxi = lxi + col * 257;

    float partial = 0.f;

    for (int it = 0; it < 4; ++it) {
        const int i0 = wave * 64 + it * 16;
        v8f accRR = {};   // Hr * xr
        v8f accII = {};   // Hi * xi
        v8f accRI = {};   // Hr * xi
        v8f accIR = {};   // Hi * xr
        const float* hr = HRk + (size_t)(i0 + col) * 256 + kk;
        const float* hi = HIk + (size_t)(i0 + col) * 256 + kk;

        for (int j0 = 0; j0 < 256; j0 += 4) {
            // A fragment (16x4, M=col, K=kk..kk+1): two contiguous f32 from H row
            v2f ar = *(const v2f*)(hr + j0);
            v2f ai = *(const v2f*)(hi + j0);
            // B fragment (4x16, K=kk..kk+1, N=col): from LDS psi tile
            v2f br, bi;
            br.x = bxr[j0 + kk]; br.y = bxr[j0 + kk + 1];
            bi.x = bxi[j0 + kk]; bi.y = bxi[j0 + kk + 1];

            accRR = __builtin_amdgcn_wmma_f32_16x16x4_f32(false, ar, false, br, (short)0, accRR, false, false);
            accII = __builtin_amdgcn_wmma_f32_16x16x4_f32(false, ai, false, bi, (short)0, accII, false, false);
            accRI = __builtin_amdgcn_wmma_f32_16x16x4_f32(false, ar, false, bi, (short)0, accRI, false, false);
            accIR = __builtin_amdgcn_wmma_f32_16x16x4_f32(false, ai, false, br, (short)0, accIR, false, false);
        }

        // y = H psi for this 16-row tile; fold into Re(psi^dag y) immediately.
        // D layout: VGPR r -> M = r (lanes 0-15) or r+8 (lanes 16-31), N = lane&15.
        const int mbase = i0 + ((lane >> 4) << 3);
        #pragma unroll
        for (int r = 0; r < 8; ++r) {
            float yr = accRR[r] - accII[r];
            float yi = accRI[r] + accIR[r];
            partial += bxr[mbase + r] * yr + bxi[mbase + r] * yi;
        }
    }

    // combine the two half-waves (rows 0-7 vs 8-15), then the 4 waves
    partial += __shfl_xor(partial, 16);
    if (lane < 16) red[wave][col] = partial;
    __syncthreads();
    if (threadIdx.x < 16) {
        float s = red[0][threadIdx.x] + red[1][threadIdx.x]
                + red[2][threadIdx.x] + red[3][threadIdx.x];
        out[(size_t)(bbase + threadIdx.x) * 64 + k] = s;
    }
}

// ---------------------------------------------------------------------------
// Launch
// ---------------------------------------------------------------------------
extern "C" void kernel_launch(void* const* d_in, const int* in_sizes, int n_in,
                              void* d_out, int out_size, void* d_ws, size_t ws_size,
                              hipStream_t stream) {
    (void)in_sizes; (void)n_in; (void)out_size; (void)ws_size;
    const float* inp = (const float*)d_in[0];  // (2048, 256)
    const float* aw  = (const float*)d_in[1];  // (2048, 194)
    const float* ano = (const float*)d_in[2];  // (64, 256, 256, 2)
    float* out = (float*)d_out;                // (2048, 64)

    float* ws = (float*)d_ws;
    float* XR = ws;                            // 2048*256
    float* XI = ws + 2048 * 256;               // 2048*256
    float* HR = ws + 2 * 2048 * 256;           // 64*256*256
    float* HI = HR + 64 * 256 * 256;           // 64*256*256

    pqc_circuit_kernel<<<2048, 256, 0, stream>>>(inp, aw, XR, XI);
    prep_H_kernel<<<(64 * 256 * 256) / 256, 256, 0, stream>>>(ano, HR, HI);
    expect_kernel<<<64 * 128, 128, 0, stream>>>(XR, XI, HR, HI, out);
}